// GCNNet_78211354460180
// MI455X (gfx1250) — compile-verified
//
#include <hip/hip_runtime.h>
#include <math.h>

// ---------------- problem constants (match reference) ----------------
#define NN      20000      // nodes (divisible by 32)
#define NE      160000     // edges
#define NG      1000       // graphs
#define IN_DIM  11
#define HID     128
#define EDGE_DIM 3
#define N_LAYERS 5
#define KDIM    (2*HID + EDGE_DIM)   // 259
#define BN_EPS  1e-5f

typedef __attribute__((ext_vector_type(2))) float v2f;
typedef __attribute__((ext_vector_type(8))) float v8f;

// ---------------- utility ----------------
__global__ void k_zero(float* p, int n) {
  for (int i = blockIdx.x * blockDim.x + threadIdx.x; i < n; i += gridDim.x * blockDim.x)
    p[i] = 0.f;
}

__global__ void k_deg(const long long* __restrict__ dst, float* __restrict__ deg) {
  int e = blockIdx.x * blockDim.x + threadIdx.x;
  if (e < NE) atomicAdd(&deg[(int)dst[e]], 1.f);
}

// h = x @ W_in + b_in   (20000 x 11 x 128 — tiny, VALU is fine)
__global__ void k_inproj(const float* __restrict__ x, const float* __restrict__ W,
                         const float* __restrict__ b, float* __restrict__ h) {
  int idx = blockIdx.x * blockDim.x + threadIdx.x;
  if (idx >= NN * HID) return;
  int i = idx >> 7, j = idx & 127;
  float acc = b[j];
#pragma unroll
  for (int k = 0; k < IN_DIM; ++k) acc += x[i * IN_DIM + k] * W[k * HID + j];
  h[idx] = acc;
}

// ---------------- heavy kernel: node-level 4-way GEMM via WMMA f32 ----------------
// Since msg = act(h[dst]@W_top + h[src]@W_mid + ea@W_edge + b), precompute the node
// products P = h @ {Wf_top, Wf_mid, Ws_top, Ws_mid} : four [NN x 128] GEMMs, K=128.
// Block = 256 thr (8 wave32), 32-node M-tile staged in LDS; each wave owns one
// 16-col N-tile and accumulates 2 M-subtiles x 4 matrices = 8 WMMAs per K-step,
// amortizing the A (LDS) and B (global, L2-resident) fragment loads 8x.
__global__ __launch_bounds__(256) void k_node_gemm(
    const float* __restrict__ h,
    const float* __restrict__ Wf, const float* __restrict__ Ws,
    float* __restrict__ Pfd, float* __restrict__ Pfs,
    float* __restrict__ Psd, float* __restrict__ Pss)
{
  __shared__ float A[32][132];             // 32 x 128 h-tile (+ pad)
  const int m0   = blockIdx.x * 32;
  const int tid  = threadIdx.x;
  const int lane = tid & 31;
  const int wave = tid >> 5;

  for (int i = tid; i < 32 * HID; i += 256) {
    int r = i >> 7, c = i & 127;
    A[r][c] = h[(m0 + r) * HID + c];
  }
  __syncthreads();

  const float* B0 = Wf;                    // Wf rows   0..127  (dst part)
  const float* B1 = Wf + 128 * HID;        // Wf rows 128..255  (src part)
  const float* B2 = Ws;
  const float* B3 = Ws + 128 * HID;

  v8f acc[2][4] = {};
  const int n0 = wave * 16;
  const int nl = lane & 15;                // M row for A-frag / N col for B-frag
  const int kh = (lane >> 4) << 1;         // lanes 0-15 -> K{0,1}, 16-31 -> K{2,3}

  for (int k0 = 0; k0 < HID; k0 += 4) {
    const int ka = k0 + kh;
    v2f a0; a0.x = A[nl][ka];      a0.y = A[nl][ka + 1];
    v2f a1; a1.x = A[16 + nl][ka]; a1.y = A[16 + nl][ka + 1];
    const int w0 = ka * HID + n0 + nl;
    const int w1 = w0 + HID;
    v2f b;
    b.x = B0[w0]; b.y = B0[w1];
    acc[0][0] = __builtin_amdgcn_wmma_f32_16x16x4_f32(false, a0, false, b, (short)0, acc[0][0], false, false);
    acc[1][0] = __builtin_amdgcn_wmma_f32_16x16x4_f32(false, a1, false, b, (short)0, acc[1][0], false, false);
    b.x = B1[w0]; b.y = B1[w1];
    acc[0][1] = __builtin_amdgcn_wmma_f32_16x16x4_f32(false, a0, false, b, (short)0, acc[0][1], false, false);
    acc[1][1] = __builtin_amdgcn_wmma_f32_16x16x4_f32(false, a1, false, b, (short)0, acc[1][1], false, false);
    b.x = B2[w0]; b.y = B2[w1];
    acc[0][2] = __builtin_amdgcn_wmma_f32_16x16x4_f32(false, a0, false, b, (short)0, acc[0][2], false, false);
    acc[1][2] = __builtin_amdgcn_wmma_f32_16x16x4_f32(false, a1, false, b, (short)0, acc[1][2], false, false);
    b.x = B3[w0]; b.y = B3[w1];
    acc[0][3] = __builtin_amdgcn_wmma_f32_16x16x4_f32(false, a0, false, b, (short)0, acc[0][3], false, false);
    acc[1][3] = __builtin_amdgcn_wmma_f32_16x16x4_f32(false, a1, false, b, (short)0, acc[1][3], false, false);
  }

  // C/D layout: VGPR r -> row (r + 8*(lane>=16)) within M-subtile, col = n0 + (lane&15)
  float* P[4] = { Pfd, Pfs, Psd, Pss };
  const int half = (lane >> 4) << 3;
  const int col  = n0 + nl;
#pragma unroll
  for (int m = 0; m < 4; ++m) {
#pragma unroll
    for (int mi = 0; mi < 2; ++mi) {
#pragma unroll
      for (int r = 0; r < 8; ++r) {
        P[m][(m0 + mi * 16 + half + r) * HID + col] = acc[mi][m][r];
      }
    }
  }
}

// Per-edge gate+softplus from precomputed node projections; scatter into agg.
// Wfe/Wse point at the 3 edge_attr rows (256..258) of the layer weights.
__global__ __launch_bounds__(256) void k_edge_combine(
    const float* __restrict__ Pfd, const float* __restrict__ Pfs,
    const float* __restrict__ Psd, const float* __restrict__ Pss,
    const long long* __restrict__ src, const long long* __restrict__ dst,
    const float* __restrict__ ea,
    const float* __restrict__ Wfe, const float* __restrict__ bf,
    const float* __restrict__ Wse, const float* __restrict__ bs,
    float* __restrict__ agg)
{
  int idx = blockIdx.x * blockDim.x + threadIdx.x;
  if (idx >= NE * HID) return;
  int e = idx >> 7, j = idx & 127;
  int d  = (int)dst[e];
  int sn = (int)src[e];
  float e0 = ea[e * EDGE_DIM + 0], e1 = ea[e * EDGE_DIM + 1], e2 = ea[e * EDGE_DIM + 2];
  float f = Pfd[d * HID + j] + Pfs[sn * HID + j]
          + e0 * Wfe[j] + e1 * Wfe[HID + j] + e2 * Wfe[2 * HID + j] + bf[j];
  float s = Psd[d * HID + j] + Pss[sn * HID + j]
          + e0 * Wse[j] + e1 * Wse[HID + j] + e2 * Wse[2 * HID + j] + bs[j];
  float sig = 1.f / (1.f + __expf(-f));
  float sp  = (s > 20.f) ? s : log1pf(__expf(s));
  atomicAdd(&agg[d * HID + j], sig * sp);
}

// h += agg/deg; accumulate per-column sum & sumsq for BN (launch: 128 blocks x 256 thr)
__global__ __launch_bounds__(256) void k_residual_stats(
    float* __restrict__ h, const float* __restrict__ agg, const float* __restrict__ deg,
    float* __restrict__ colsum, float* __restrict__ colsq)
{
  const int col = threadIdx.x & 127;
  const int sub = threadIdx.x >> 7;
  float s = 0.f, q = 0.f;
  for (int row = blockIdx.x * 2 + sub; row < NN; row += gridDim.x * 2) {
    int idx = row * HID + col;
    float v = h[idx] + agg[idx] / fmaxf(deg[row], 1.f);
    h[idx] = v;
    s += v; q += v * v;
  }
  __shared__ float sh[256], shq[256];
  sh[threadIdx.x] = s; shq[threadIdx.x] = q;
  __syncthreads();
  if (sub == 0) {
    s += sh[col + 128]; q += shq[col + 128];
    atomicAdd(&colsum[col], s);
    atomicAdd(&colsq[col], q);
  }
}

__global__ void k_bn_finalize(const float* __restrict__ colsum, const float* __restrict__ colsq,
                              float* __restrict__ mu, float* __restrict__ rsig) {
  int j = threadIdx.x;                 // 128 threads
  float m = colsum[j] * (1.f / NN);
  float v = colsq[j] * (1.f / NN) - m * m;
  mu[j] = m;
  rsig[j] = rsqrtf(v + BN_EPS);
}

__global__ void k_bn_apply(float* __restrict__ h, const float* __restrict__ mu,
                           const float* __restrict__ rsig, const float* __restrict__ gamma,
                           const float* __restrict__ beta) {
  int idx = blockIdx.x * blockDim.x + threadIdx.x;
  if (idx >= NN * HID) return;
  int j = idx & 127;
  float v = (h[idx] - mu[j]) * rsig[j] * gamma[j] + beta[j];
  h[idx] = v > 0.f ? v : 0.f;
}

// ---------------- pooling + head ----------------
__global__ void k_pool(const float* __restrict__ h, const long long* __restrict__ batch,
                       float* __restrict__ gsum, float* __restrict__ gcnt) {
  int idx = blockIdx.x * blockDim.x + threadIdx.x;
  if (idx >= NN * HID) return;
  int i = idx >> 7, j = idx & 127;
  int b = (int)batch[i];
  atomicAdd(&gsum[b * HID + j], h[idx]);
  if (j == 0) atomicAdd(&gcnt[b], 1.f);
}

__global__ void k_pool_div(float* __restrict__ gsum, const float* __restrict__ gcnt) {
  int idx = blockIdx.x * blockDim.x + threadIdx.x;
  if (idx >= NG * HID) return;
  gsum[idx] /= fmaxf(gcnt[idx >> 7], 1.f);
}

__global__ void k_dense_relu(const float* __restrict__ in, const float* __restrict__ W,
                             const float* __restrict__ b, float* __restrict__ out) {
  int idx = blockIdx.x * blockDim.x + threadIdx.x;
  if (idx >= NG * HID) return;
  int i = idx >> 7, j = idx & 127;
  float acc = b[j];
#pragma unroll 8
  for (int k = 0; k < HID; ++k) acc += in[i * HID + k] * W[k * HID + j];
  out[idx] = acc > 0.f ? acc : 0.f;
}

__global__ void k_dense_out(const float* __restrict__ in, const float* __restrict__ W3,
                            const float* __restrict__ b3, float* __restrict__ out) {
  int i = blockIdx.x * blockDim.x + threadIdx.x;
  if (i >= NG) return;
  float acc = b3[0];
#pragma unroll 8
  for (int k = 0; k < HID; ++k) acc += in[i * HID + k] * W3[k];
  out[i] = acc;
}

// ---------------- launcher ----------------
extern "C" void kernel_launch(void* const* d_in, const int* in_sizes, int n_in,
                              void* d_out, int out_size, void* d_ws, size_t ws_size,
                              hipStream_t stream) {
  const float*     x    = (const float*)d_in[0];
  const long long* ei   = (const long long*)d_in[1];   // int64 [2, NE]
  const long long* batc = (const long long*)d_in[2];   // int64 [NN]
  const float*     ea   = (const float*)d_in[3];
  const float*     W_in = (const float*)d_in[4];
  const float*     b_in = (const float*)d_in[5];
  const float*     Wf   = (const float*)d_in[6];       // [5, 259, 128]
  const float*     bf   = (const float*)d_in[7];
  const float*     Ws   = (const float*)d_in[8];
  const float*     bs   = (const float*)d_in[9];
  const float*     gam  = (const float*)d_in[10];
  const float*     bet  = (const float*)d_in[11];
  const float*     W1   = (const float*)d_in[12];
  const float*     b1   = (const float*)d_in[13];
  const float*     W2   = (const float*)d_in[14];
  const float*     b2   = (const float*)d_in[15];
  const float*     W3   = (const float*)d_in[16];
  const float*     b3   = (const float*)d_in[17];
  float*           outp = (float*)d_out;

  const long long* srcI = ei;        // edge_index[0]
  const long long* dstI = ei + NE;   // edge_index[1]

  // workspace layout (floats); agg|colsum|colsq contiguous so one zero covers them
  float* ws     = (float*)d_ws;
  float* h      = ws;
  float* agg    = h      + (size_t)NN * HID;
  float* colsum = agg    + (size_t)NN * HID;
  float* colsq  = colsum + HID;
  float* deg    = colsq  + HID;
  float* mu     = deg    + NN;
  float* rsig   = mu     + HID;
  float* gsum   = rsig   + HID;
  float* gcnt   = gsum   + (size_t)NG * HID;
  float* g1     = gcnt   + NG;
  float* g2     = g1     + (size_t)NG * HID;
  float* Pfd    = g2     + (size_t)NG * HID;
  float* Pfs    = Pfd    + (size_t)NN * HID;
  float* Psd    = Pfs    + (size_t)NN * HID;
  float* Pss    = Psd    + (size_t)NN * HID;

  const int T  = 256;
  const int nh = (NN * HID + T - 1) / T;

  k_zero<<<256, T, 0, stream>>>(deg, NN);
  k_deg<<<(NE + T - 1) / T, T, 0, stream>>>(dstI, deg);
  k_inproj<<<nh, T, 0, stream>>>(x, W_in, b_in, h);

  for (int l = 0; l < N_LAYERS; ++l) {
    const float* Wf_l = Wf + (size_t)l * KDIM * HID;
    const float* Ws_l = Ws + (size_t)l * KDIM * HID;
    k_node_gemm<<<NN / 32, T, 0, stream>>>(h, Wf_l, Ws_l, Pfd, Pfs, Psd, Pss);
    k_zero<<<512, T, 0, stream>>>(agg, NN * HID + 2 * HID);   // agg + colsum + colsq
    k_edge_combine<<<(NE * HID) / T, T, 0, stream>>>(
        Pfd, Pfs, Psd, Pss, srcI, dstI, ea,
        Wf_l + 2 * HID * HID, bf + l * HID,
        Ws_l + 2 * HID * HID, bs + l * HID, agg);
    k_residual_stats<<<128, T, 0, stream>>>(h, agg, deg, colsum, colsq);
    k_bn_finalize<<<1, HID, 0, stream>>>(colsum, colsq, mu, rsig);
    k_bn_apply<<<nh, T, 0, stream>>>(h, mu, rsig, gam + l * HID, bet + l * HID);
  }

  k_zero<<<128, T, 0, stream>>>(gsum, NG * HID + NG);         // gsum + gcnt
  k_pool<<<nh, T, 0, stream>>>(h, batc, gsum, gcnt);
  k_pool_div<<<(NG * HID + T - 1) / T, T, 0, stream>>>(gsum, gcnt);
  k_dense_relu<<<(NG * HID + T - 1) / T, T, 0, stream>>>(gsum, W1, b1, g1);
  k_dense_relu<<<(NG * HID + T - 1) / T, T, 0, stream>>>(g1, W2, b2, g2);
  k_dense_out<<<(NG + T - 1) / T, T, 0, stream>>>(g2, W3, b3, outp);
}